// Detector_78400333021669
// MI455X (gfx1250) — compile-verified
//
#include <hip/hip_runtime.h>
#include <hip/hip_bf16.h>
#include <cstdint>

#define EPSF     1e-7f
#define POS_THR  0.7f
#define NEG_THR  0.3f
#define MM       128   // labels per batch
#define CC       80    // classes
#define BB_      16    // batch

typedef unsigned int u32;
typedef u32 __attribute__((ext_vector_type(4))) v4u;
typedef int  __attribute__((ext_vector_type(4))) v4i;
typedef int  __attribute__((ext_vector_type(8))) v8i;

__device__ __forceinline__ float wave_sum(float v) {
    // wave32 butterfly reduction
    v += __shfl_down(v, 16, 32);
    v += __shfl_down(v,  8, 32);
    v += __shfl_down(v,  4, 32);
    v += __shfl_down(v,  2, 32);
    v += __shfl_down(v,  1, 32);
    return v;
}

// a/d via v_rcp_f32 + one Newton-Raphson step (~1 ulp): 4 VALU ops instead of
// the ~10-op IEEE divide expansion. Used only in the B*N*M-iteration (~51.6M) hot loop.
__device__ __forceinline__ float fast_div(float a, float d) {
    float r = __builtin_amdgcn_rcpf(d);
    float e = __builtin_fmaf(-d, r, 1.0f);
    r = __builtin_fmaf(r, e, r);
    return a * r;
}

__global__ void det_zero_ws(float* acc) {
    if (threadIdx.x < 6) acc[threadIdx.x] = 0.0f;
}

__global__ __launch_bounds__(256) void det_main(
    const float* __restrict__ label_box, const float* __restrict__ box,
    const float* __restrict__ prb, const float* __restrict__ cls,
    const int* __restrict__ label_cls, float* __restrict__ acc, int N)
{
    __shared__ float4 s_lbox[MM];     // 2 KB label-box tile (TDM target)
    __shared__ float  s_area[MM];
    __shared__ int    s_lcls[MM];
    __shared__ float  s_acc[6];

    const int b = blockIdx.y;
    const int n = blockIdx.x * blockDim.x + threadIdx.x;

    // ---- Stage label_box[b] (128 x 4 f32 = 2KB) into LDS via the TDM ----
#if __has_builtin(__builtin_amdgcn_tensor_load_to_lds)
    if (threadIdx.x < 32) {   // wave 0 only issues the TDM descriptor
        unsigned long long ga =
            (unsigned long long)(uintptr_t)(label_box + (size_t)b * MM * 4);
        u32 lds_off = (u32)(uintptr_t)(void*)&s_lbox[0]; // low 32 bits = LDS byte offset
        // D# group0: count=1, lds_addr, global_addr[56:0], type=2 ("image")
        v4u g0 = { 1u,
                   lds_off,
                   (u32)ga,
                   ((u32)(ga >> 32) & 0x01FFFFFFu) | (2u << 30) };
        // D# group1: data_size=4B; 1D tensor/tile of 512 elements
        v8i g1 = { (int)(2u << 16),     // data_size=2 (4 bytes), wg_mask=0
                   (int)(512u << 16),   // tensor_dim0 = 512 (low 16 bits)
                   (int)(1u  << 16),    // tensor_dim0 hi=0 | tensor_dim1 = 1
                   (int)(512u << 16),   // tensor_dim1 hi=0 | tile_dim0 = 512
                   1,                   // tile_dim1 = 1, tile_dim2 = 0
                   512,                 // tensor_dim0_stride = 512
                   (int)(512u << 16),   // stride0 hi=0 | tensor_dim1_stride = 512
                   0 };
        v4i z4 = {0, 0, 0, 0};          // groups 2/3: dims unused (<=2D)
        v8i z8 = {0, 0, 0, 0, 0, 0, 0, 0};
        __builtin_amdgcn_tensor_load_to_lds(g0, g1, z4, z4, z8, 0); // 6-arg form
        __builtin_amdgcn_s_wait_tensorcnt(0);  // s_wait_tensorcnt 0
    }
#else
    for (int i = threadIdx.x; i < MM * 4; i += blockDim.x)
        ((float*)s_lbox)[i] = label_box[(size_t)b * MM * 4 + i];
#endif
    for (int m = threadIdx.x; m < MM; m += blockDim.x)
        s_lcls[m] = label_cls[b * MM + m];
    if (threadIdx.x < 6) s_acc[threadIdx.x] = 0.0f;
    __syncthreads();

    // Precompute per-label area. NOTE: the reference's validity mask is a
    // mathematical no-op for the IoU values: an invalid label is the literal
    // box [0,0,0,0], whose clamped intersection with any box is exactly 0,
    // so its IoU is already 0 without masking. We therefore skip it.
    for (int m = threadIdx.x; m < MM; m += blockDim.x) {
        float4 a = s_lbox[m];
        s_area[m] = fmaxf(a.z - a.x, 0.0f) * fmaxf(a.w - a.y, 0.0f);
    }
    __syncthreads();

    float c_pos = 0.f, c_neg = 0.f, s_box = 0.f, s_logp = 0.f,
          s_log1mp = 0.f, s_nll = 0.f;

    if (n < N) {
        const size_t bn = (size_t)b * N + n;
        // Pre-pull the cls gather row (emits global_prefetch_b8)
        __builtin_prefetch((const void*)(cls + bn * CC), 0, 3);

        const float4 bb = ((const float4*)box)[bn];       // global_load_b128
        const float area_b = fmaxf(bb.z - bb.x, 0.0f) * fmaxf(bb.w - bb.y, 0.0f);

        float best = 0.0f;   // IoUs are >= 0; argmax of all-zero row = 0
        int   besti = 0;
        #pragma unroll 8
        for (int m = 0; m < MM; ++m) {
            float4 a = s_lbox[m];                          // LDS broadcast b128
            float ix1 = fmaxf(a.x, bb.x);
            float iy1 = fmaxf(a.y, bb.y);
            float ix2 = fminf(a.z, bb.z);
            float iy2 = fminf(a.w, bb.w);
            float inter = fmaxf(ix2 - ix1, 0.0f) * fmaxf(iy2 - iy1, 0.0f);
            float uni   = s_area[m] + area_b - inter;
            float iou   = fast_div(inter, uni + EPSF);
            if (iou > best) { best = iou; besti = m; }     // strict > = first max
        }

        const bool pos = best > POS_THR;
        const bool neg = (best > 0.0f) && (best < NEG_THR);
        const float p = fminf(fmaxf(prb[bn], EPSF), 1.0f - EPSF);

        if (pos) {
            c_pos  = 1.0f;
            s_logp = -logf(p);
            // ---- CIoU(detect_box, box) exactly per reference ----
            float4 a = s_lbox[besti];
            float ix1 = fmaxf(a.x, bb.x), iy1 = fmaxf(a.y, bb.y);
            float ix2 = fminf(a.z, bb.z), iy2 = fminf(a.w, bb.w);
            float inter = fmaxf(ix2 - ix1, 0.0f) * fmaxf(iy2 - iy1, 0.0f);
            float w1 = fmaxf(a.z - a.x, 0.0f),  h1 = fmaxf(a.w - a.y, 0.0f);
            float w2 = fmaxf(bb.z - bb.x, 0.0f), h2 = fmaxf(bb.w - bb.y, 0.0f);
            float uni = w1 * h1 + w2 * h2 - inter;
            float iou = inter / (uni + EPSF);
            float cw  = fmaxf(a.z, bb.z) - fminf(a.x, bb.x);
            float ch  = fmaxf(a.w, bb.w) - fminf(a.y, bb.y);
            float c2  = cw * cw + ch * ch + EPSF;
            float dx  = (a.x + a.z - bb.x - bb.z) * 0.5f;
            float dy  = (a.y + a.w - bb.y - bb.w) * 0.5f;
            float rho2 = dx * dx + dy * dy;
            float dat = atanf(w1 / (h1 + EPSF)) - atanf(w2 / (h2 + EPSF));
            float v   = 0.40528473456935109f * dat * dat;   // 4/pi^2
            float alpha = v / (1.0f - iou + v + EPSF);
            s_box = 1.0f - (iou - rho2 / c2 - alpha * v);
            // ---- class NLL on gathered class prob ----
            int dc = s_lcls[besti];
            float pc = fminf(fmaxf(cls[bn * CC + dc], EPSF), 1.0f);
            s_nll = -logf(pc);
        } else if (neg) {
            c_neg    = 1.0f;
            s_log1mp = -logf(1.0f - p);
        }
    }

    // wave32 shuffle reduce -> LDS atomics -> global atomics
    c_pos = wave_sum(c_pos);   c_neg    = wave_sum(c_neg);
    s_box = wave_sum(s_box);   s_logp   = wave_sum(s_logp);
    s_log1mp = wave_sum(s_log1mp); s_nll = wave_sum(s_nll);
    if ((threadIdx.x & 31) == 0) {
        atomicAdd(&s_acc[0], c_pos);
        atomicAdd(&s_acc[1], c_neg);
        atomicAdd(&s_acc[2], s_box);
        atomicAdd(&s_acc[3], s_logp);
        atomicAdd(&s_acc[4], s_log1mp);
        atomicAdd(&s_acc[5], s_nll);
    }
    __syncthreads();
    if (threadIdx.x < 6) atomicAdd(&acc[threadIdx.x], s_acc[threadIdx.x]);
}

__global__ void det_finalize(const float* __restrict__ acc, float* __restrict__ out) {
    if (threadIdx.x == 0) {
        float cpos = fmaxf(acc[0], 1.0f);
        float cneg = fmaxf(acc[1], 1.0f);
        out[0] = acc[2] / cpos;                    // box_loss
        out[1] = acc[3] / cpos + acc[4] / cneg;    // prb_loss
        out[2] = acc[5] / cpos;                    // cls_loss
    }
}

extern "C" void kernel_launch(void* const* d_in, const int* in_sizes, int n_in,
                              void* d_out, int out_size, void* d_ws, size_t ws_size,
                              hipStream_t stream) {
    const float* label_box = (const float*)d_in[0];   // (B, M, 4) f32
    const float* box       = (const float*)d_in[1];   // (B, N, 4) f32
    const float* prb       = (const float*)d_in[2];   // (B, N)    f32
    const float* cls       = (const float*)d_in[3];   // (B, N, C) f32
    const int*   label_cls = (const int*)  d_in[4];   // (B, M)    i32
    float* out = (float*)d_out;                       // 3 f32
    float* acc = (float*)d_ws;                        // 6 f32 accumulators

    const int N = in_sizes[2] / BB_;                  // 25200

    det_zero_ws<<<1, 32, 0, stream>>>(acc);
    dim3 grid((N + 255) / 256, BB_);
    det_main<<<grid, 256, 0, stream>>>(label_box, box, prb, cls, label_cls, acc, N);
    det_finalize<<<1, 32, 0, stream>>>(acc, out);
}